// ModelDisigner_45028437131548
// MI455X (gfx1250) — compile-verified
//
#include <hip/hip_runtime.h>
#include <hip/hip_bf16.h>
#include <math.h>

// ---------------------------------------------------------------------------
// MI455X (gfx1250, wave32). Matrix work uses V_WMMA_F32_16X16X4_F32 so the
// f32 reference numerics are preserved while the matrix pipe is exercised.
// Each wave computes a 64(m) x 16(n) strip: 4 accumulators share one B
// fragment, amortizing the im2col address math and B loads 4x.
// ---------------------------------------------------------------------------

typedef __attribute__((ext_vector_type(2))) float v2f;
typedef __attribute__((ext_vector_type(8))) float v8f;

__device__ __forceinline__ v8f wmma_f32_k4(v2f a, v2f b, v8f c) {
  // (neg_a, A, neg_b, B, c_mod, C, reuse_a, reuse_b)
  return __builtin_amdgcn_wmma_f32_16x16x4_f32(
      false, a, false, b, (short)0, c, false, false);
}

// ---------------------------------------------------------------------------
// Kernel 1: direct 3x3 stride-2 conv, Cin=3 -> Cout=64, + ReLU
// ---------------------------------------------------------------------------
__global__ void conv1_kernel(const float* __restrict__ X,
                             const float* __restrict__ W,
                             const float* __restrict__ bias,
                             float* __restrict__ Y,
                             int Bn, int Hin, int Win, int Hout, int Wout) {
  long idx = (long)blockIdx.x * blockDim.x + threadIdx.x;
  long total = (long)Bn * 64 * Hout * Wout;
  if (idx >= total) return;
  int ox = (int)(idx % Wout);
  long t = idx / Wout;
  int oy = (int)(t % Hout);  t /= Hout;
  int co = (int)(t % 64);
  int b  = (int)(t / 64);

  const float* Xb = X + ((long)b * 3 * Hin + (long)oy * 2) * Win + ox * 2;
  const float* Wr = W + co * 27;
  int inHW = Hin * Win;
  float acc = bias[co];
#pragma unroll
  for (int ci = 0; ci < 3; ++ci)
#pragma unroll
    for (int ky = 0; ky < 3; ++ky)
#pragma unroll
      for (int kx = 0; kx < 3; ++kx)
        acc += Wr[ci * 9 + ky * 3 + kx] * Xb[ci * inHW + ky * Win + kx];
  Y[idx] = fmaxf(acc, 0.0f);
}

// ---------------------------------------------------------------------------
// Kernel 2: implicit-GEMM 3x3 stride-2 conv via f32 WMMA (16x16x4).
// Wave computes 64(co) x 16(n); n = (b,oy,ox) flattened. im2col k-offsets are
// precomputed into an LDS table (one ds_load_b64 per K-step replaces the
// div/mod decode). Fragment layouts per CDNA5 ISA 7.12.2:
//   A/B: lane&15 = row/col, k = 2*(lane>>4)+v ; C/D: m = v + 8*(lane>>4).
// ---------------------------------------------------------------------------
__global__ void conv3x3s2_wmma(const float* __restrict__ X,
                               const float* __restrict__ W,
                               const float* __restrict__ bias,
                               float* __restrict__ Y,
                               int Bn, int Cin, int Hin, int Win,
                               int Cout, int Hout, int Wout, int relu) {
  __shared__ int offs[1152];                    // max K = 128*9
  const int K  = Cin * 9;
  const int Nn = Bn * Hout * Wout;
  const int nTilesN = (Nn + 15) >> 4;
  const int mGroups = Cout >> 6;                // 4 m-tiles (64 ch) per wave
  const int numTiles = mGroups * nTilesN;
  const int inHW = Hin * Win;

  // Cooperative im2col offset table (before any wave exits).
  for (int k = threadIdx.x; k < K; k += blockDim.x) {
    int ci = k / 9, r = k - ci * 9, ky = r / 3, kx = r - ky * 3;
    offs[k] = ci * inHW + ky * Win + kx;
  }
  __syncthreads();

  int wave = blockIdx.x * (blockDim.x >> 5) + ((int)threadIdx.x >> 5);
  if (wave >= numTiles) return;                 // wave-uniform exit

  int mG    = wave / nTilesN;
  int nTile = wave - mG * nTilesN;
  int lane = (int)threadIdx.x & 31;
  int half = lane >> 4;                         // 0 or 1
  int l16  = lane & 15;

  // B column (n) decode, clamped so EXEC stays all-ones for WMMA.
  int n = (nTile << 4) + l16;
  int valid = (n < Nn);
  int nc = valid ? n : (Nn - 1);
  int HW = Hout * Wout;
  int bidx = nc / HW;
  int rem  = nc - bidx * HW;
  int oy   = rem / Wout;
  int ox   = rem - oy * Wout;
  const float* Xb = X + ((long)bidx * Cin * Hin + (long)oy * 2) * Win + ox * 2;

  // A rows: co = mG*64 + l16 + {0,16,32,48}
  const float* Wr = W + (long)(mG * 64 + l16) * K;
  const long WK = (long)16 * K;

  v8f acc0 = {0.f,0.f,0.f,0.f,0.f,0.f,0.f,0.f};
  v8f acc1 = acc0, acc2 = acc0, acc3 = acc0;
  const int koff = half * 2;

  for (int k0 = 0; k0 < K; k0 += 4) {
    int k = k0 + koff;
    int o0 = offs[k], o1 = offs[k + 1];         // ds_load_b64
    v2f b;  b.x = Xb[o0];            b.y = Xb[o1];
    v2f a0; a0.x = Wr[k];            a0.y = Wr[k + 1];
    v2f a1; a1.x = Wr[WK + k];       a1.y = Wr[WK + k + 1];
    v2f a2; a2.x = Wr[2 * WK + k];   a2.y = Wr[2 * WK + k + 1];
    v2f a3; a3.x = Wr[3 * WK + k];   a3.y = Wr[3 * WK + k + 1];
    acc0 = wmma_f32_k4(a0, b, acc0);
    acc1 = wmma_f32_k4(a1, b, acc1);
    acc2 = wmma_f32_k4(a2, b, acc2);
    acc3 = wmma_f32_k4(a3, b, acc3);
  }

  if (valid) {
    long obase = (long)bidx * Cout * HW + (long)oy * Wout + ox;
    v8f accs[4] = {acc0, acc1, acc2, acc3};
#pragma unroll
    for (int t = 0; t < 4; ++t)
#pragma unroll
      for (int v = 0; v < 8; ++v) {
        int m = mG * 64 + t * 16 + v + half * 8;
        float val = accs[t][v] + bias[m];
        if (relu) val = fmaxf(val, 0.0f);
        Y[obase + (long)m * HW] = val;
      }
  }
}

// ---------------------------------------------------------------------------
// Kernel 3: depthwise cross-correlation. Block per (b,c): 31x31 (x) 15x15 -> 17x17.
// ---------------------------------------------------------------------------
__global__ void corr_kernel(const float* __restrict__ sf,
                            const float* __restrict__ tf,
                            float* __restrict__ corr) {
  __shared__ float s[961];
  __shared__ float t[225];
  int bc = blockIdx.x;
  const float* sp = sf + (long)bc * 961;
  const float* tp = tf + (long)bc * 225;
  for (int i = threadIdx.x; i < 961; i += blockDim.x) s[i] = sp[i];
  for (int i = threadIdx.x; i < 225; i += blockDim.x) t[i] = tp[i];
  __syncthreads();
  int p = threadIdx.x;
  if (p < 289) {
    int py = p / 17, px = p - py * 17;
    float acc = 0.0f;
#pragma unroll 3
    for (int ky = 0; ky < 15; ++ky)
#pragma unroll 5
      for (int kx = 0; kx < 15; ++kx)
        acc += s[(py + ky) * 31 + px + kx] * t[ky * 15 + kx];
    corr[(long)bc * 289 + p] = acc;
  }
}

// ---------------------------------------------------------------------------
// Kernel 4: score (1x1 conv 256->1 + sigmoid), first-max argmax, raw-reshape
// row gather: row[b,k] = corr_flat[b, 4352*r + 256*c + k].
// ---------------------------------------------------------------------------
__global__ void score_argmax_gather(const float* __restrict__ corr,
                                    const float* __restrict__ w_s,
                                    const float* __restrict__ b_s,
                                    float* __restrict__ score_out,
                                    float* __restrict__ rowbuf) {
  __shared__ float zbuf[289];
  __shared__ int best;
  int b = blockIdx.x;
  int t = threadIdx.x;
  const float* cb = corr + (long)b * 73984;     // 256*289
  if (t < 289) {
    float z = b_s[0];
    for (int c = 0; c < 256; ++c) z += w_s[c] * cb[c * 289 + t];
    zbuf[t] = z;
    score_out[b * 289 + t] = 1.0f / (1.0f + expf(-z));  // sigmoid (monotone)
  }
  __syncthreads();
  if (t == 0) {
    float bz = zbuf[0];
    int bi = 0;
    for (int i = 1; i < 289; ++i)
      if (zbuf[i] > bz) { bz = zbuf[i]; bi = i; }        // first-max tie-break
    best = bi;
  }
  __syncthreads();
  int r = best / 17, c = best - r * 17;
  int off = 4352 * r + 256 * c;                 // raw-reshape gather offset
  if (t < 256) rowbuf[b * 256 + t] = cb[off + t];
}

// ---------------------------------------------------------------------------
// Kernel 5: dense f32 WMMA GEMM, C[n,m] = sum_k A[m,k]*Bm[n,k] + bias[m].
// A row-major [M,K] (1x1-conv weights), Bm row-major [N,K]. Wave = 64m x 16n.
// ---------------------------------------------------------------------------
__global__ void gemm_wmma(const float* __restrict__ A,
                          const float* __restrict__ Bm,
                          const float* __restrict__ bias,
                          float* __restrict__ C,
                          int M, int Nn, int K, int relu) {
  int nTilesN = (Nn + 15) >> 4;
  int mGroups = M >> 6;                         // M divisible by 64 (512, 4096)
  int numTiles = mGroups * nTilesN;
  int wave = blockIdx.x * (blockDim.x >> 5) + ((int)threadIdx.x >> 5);
  if (wave >= numTiles) return;
  int mG    = wave / nTilesN;
  int nTile = wave - mG * nTilesN;
  int lane = (int)threadIdx.x & 31;
  int half = lane >> 4, l16 = lane & 15;

  const float* Ar = A + (long)(mG * 64 + l16) * K;
  const long AK = (long)16 * K;
  int nrow = (nTile << 4) + l16;
  int nclamp = nrow < Nn ? nrow : (Nn - 1);
  const float* Br = Bm + (long)nclamp * K;

  v8f acc0 = {0.f,0.f,0.f,0.f,0.f,0.f,0.f,0.f};
  v8f acc1 = acc0, acc2 = acc0, acc3 = acc0;
  const int koff = half * 2;
  for (int k0 = 0; k0 < K; k0 += 4) {
    int k = k0 + koff;
    v2f b;  b.x  = Br[k];            b.y  = Br[k + 1];
    v2f a0; a0.x = Ar[k];            a0.y = Ar[k + 1];
    v2f a1; a1.x = Ar[AK + k];       a1.y = Ar[AK + k + 1];
    v2f a2; a2.x = Ar[2 * AK + k];   a2.y = Ar[2 * AK + k + 1];
    v2f a3; a3.x = Ar[3 * AK + k];   a3.y = Ar[3 * AK + k + 1];
    acc0 = wmma_f32_k4(a0, b, acc0);
    acc1 = wmma_f32_k4(a1, b, acc1);
    acc2 = wmma_f32_k4(a2, b, acc2);
    acc3 = wmma_f32_k4(a3, b, acc3);
  }
  if (nrow < Nn) {
    v8f accs[4] = {acc0, acc1, acc2, acc3};
#pragma unroll
    for (int t = 0; t < 4; ++t)
#pragma unroll
      for (int v = 0; v < 8; ++v) {
        int m = mG * 64 + t * 16 + v + half * 8;
        float val = accs[t][v] + bias[m];
        if (relu) val = fmaxf(val, 0.0f);
        C[(long)nrow * M + m] = val;
      }
  }
}

// ---------------------------------------------------------------------------
// Kernel 6: 2x nearest upsample + scalar 1x1 conv + sigmoid -> [B,1,128,128]
// ---------------------------------------------------------------------------
__global__ void final_kernel(const float* __restrict__ m,
                             const float* __restrict__ wf,
                             const float* __restrict__ bf,
                             float* __restrict__ out) {
  int idx = blockIdx.x * blockDim.x + threadIdx.x;
  if (idx >= 32 * 128 * 128) return;
  int x = idx & 127;
  int y = (idx >> 7) & 127;
  int b = idx >> 14;
  float v = m[b * 4096 + (y >> 1) * 64 + (x >> 1)];
  float z = wf[0] * v + bf[0];
  out[idx] = 1.0f / (1.0f + expf(-z));
}

// ---------------------------------------------------------------------------
// Host-side orchestration
// ---------------------------------------------------------------------------
extern "C" void kernel_launch(void* const* d_in, const int* in_sizes, int n_in,
                              void* d_out, int out_size, void* d_ws, size_t ws_size,
                              hipStream_t stream) {
  const float* target  = (const float*)d_in[0];   // [32,3,127,127]
  const float* searchs = (const float*)d_in[1];   // [32,3,255,255]
  const float* w1 = (const float*)d_in[2];  const float* b1 = (const float*)d_in[3];
  const float* w2 = (const float*)d_in[4];  const float* b2 = (const float*)d_in[5];
  const float* w3 = (const float*)d_in[6];  const float* b3 = (const float*)d_in[7];
  const float* w_s = (const float*)d_in[8]; const float* b_s = (const float*)d_in[9];
  const float* wm1 = (const float*)d_in[10]; const float* bm1 = (const float*)d_in[11];
  const float* wm2 = (const float*)d_in[12]; const float* bm2 = (const float*)d_in[13];
  const float* wf = (const float*)d_in[14]; const float* bf = (const float*)d_in[15];
  float* out = (float*)d_out;
  float* wsp = (float*)d_ws;

  // Workspace layout (floats). t3/s3/corr/etc reuse the dead conv1 region.
  const long T1 = 0;                       // 32*64*63*63   = 8,128,512
  const long S1 = 8128512;                 // 32*64*127*127 = 33,032,192
  const long T2 = 41160704;                // 32*128*31*31  = 3,936,256
  const long S2 = 45096960;                // 32*128*63*63  = 16,257,024
  const long T3 = 0;                       // 32*256*15*15  = 1,843,200
  const long S3 = 1843200;                 // 32*256*31*31  = 7,872,512
  const long CR = 9715712;                 // 32*256*289    = 2,367,488
  const long RW = 12083200;                // 32*256
  const long HB = 12091392;                // 32*512
  const long MB = 12107776;                // 32*4096

  auto cdiv = [](long a, long b) { return (a + b - 1) / b; };

  // --- backbone layer 1 (direct) ---
  {
    long n = 32L * 64 * 63 * 63;
    conv1_kernel<<<(unsigned)cdiv(n, 256), 256, 0, stream>>>(
        target, w1, b1, wsp + T1, 32, 127, 127, 63, 63);
    n = 32L * 64 * 127 * 127;
    conv1_kernel<<<(unsigned)cdiv(n, 256), 256, 0, stream>>>(
        searchs, w1, b1, wsp + S1, 32, 255, 255, 127, 127);
  }

  // --- backbone layers 2 & 3 (implicit GEMM, f32 WMMA, 64x16 per wave) ---
  auto launch_conv = [&](const float* X, const float* W, const float* bi, float* Y,
                         int Cin, int Hin, int Win, int Cout, int Hout, int Wout) {
    long Nn = 32L * Hout * Wout;
    long tiles = (long)(Cout / 64) * ((Nn + 15) / 16);
    conv3x3s2_wmma<<<(unsigned)cdiv(tiles, 4), 128, 0, stream>>>(
        X, W, bi, Y, 32, Cin, Hin, Win, Cout, Hout, Wout, 1);
  };
  launch_conv(wsp + T1, w2, b2, wsp + T2,  64,  63,  63, 128, 31, 31);
  launch_conv(wsp + S1, w2, b2, wsp + S2,  64, 127, 127, 128, 63, 63);
  launch_conv(wsp + T2, w3, b3, wsp + T3, 128,  31,  31, 256, 15, 15);
  launch_conv(wsp + S2, w3, b3, wsp + S3, 128,  63,  63, 256, 31, 31);

  // --- depthwise cross-correlation: 8192 (b,c) pairs ---
  corr_kernel<<<32 * 256, 320, 0, stream>>>(wsp + S3, wsp + T3, wsp + CR);

  // --- score + argmax + raw-reshape row gather ---
  score_argmax_gather<<<32, 320, 0, stream>>>(wsp + CR, w_s, b_s, out, wsp + RW);

  // --- mask branch GEMMs (f32 WMMA): 512x256 relu, then 4096x512 ---
  {
    long t1 = (512L / 64) * 2;    // N=32 -> 2 n-tiles
    gemm_wmma<<<(unsigned)cdiv(t1, 4), 128, 0, stream>>>(
        wm1, wsp + RW, bm1, wsp + HB, 512, 32, 256, 1);
    long t2 = (4096L / 64) * 2;
    gemm_wmma<<<(unsigned)cdiv(t2, 4), 128, 0, stream>>>(
        wm2, wsp + HB, bm2, wsp + MB, 4096, 32, 512, 0);
  }

  // --- upsample + final scalar conv + sigmoid ---
  final_kernel<<<(unsigned)cdiv(32L * 128 * 128, 256), 256, 0, stream>>>(
      wsp + MB, wf, bf, out + 32 * 289);
}